// TopoLoss2_16406775071322
// MI455X (gfx1250) — compile-verified
//
#include <hip/hip_runtime.h>

#define BATCH 16
#define NPTS  2048
#define NDIM  64
#define NEDGE (NPTS - 1)

typedef __attribute__((ext_vector_type(2))) float v2f;
typedef __attribute__((ext_vector_type(8))) float v8f;

// ---------------------------------------------------------------------------
// Kernel 1: squared norms  sq[b*N + i] = sum_d X[b,i,d]^2
// ---------------------------------------------------------------------------
__global__ __launch_bounds__(256) void sqnorm_kernel(const float* __restrict__ X,
                                                     float* __restrict__ sq) {
    int p = blockIdx.x * 256 + threadIdx.x;
    if (p >= BATCH * NPTS) return;
    const float* x = X + (size_t)p * NDIM;
    float s = 0.0f;
#pragma unroll
    for (int d = 0; d < NDIM; d += 4) {
        float4 v = *(const float4*)(x + d);
        s += v.x * v.x + v.y * v.y + v.z * v.z + v.w * v.w;
    }
    sq[p] = s;
}

// ---------------------------------------------------------------------------
// Kernel 2: pairwise distance via WMMA Gram matrix.
// One wave per 16x16 output tile; K=64 swept with 16 x V_WMMA_F32_16X16X4_F32.
// D[b,i,j] = sqrt(max(sq[i] + sq[j] - 2*(X_i . X_j), 0))
// ---------------------------------------------------------------------------
__global__ __launch_bounds__(256) void gram_dist_kernel(const float* __restrict__ X,
                                                        const float* __restrict__ sq,
                                                        float* __restrict__ D) {
    const int tilesPerRow   = NPTS / 16;                 // 128
    const int tilesPerBatch = tilesPerRow * tilesPerRow; // 16384
    int wave = (blockIdx.x * 256 + threadIdx.x) >> 5;
    int lane = threadIdx.x & 31;
    int b  = wave / tilesPerBatch;
    int t  = wave % tilesPerBatch;
    int ti = t / tilesPerRow;
    int tj = t % tilesPerRow;

    int r = lane & 15;   // row/col within tile handled by this lane
    int h = lane >> 4;   // half-wave selects K pair / output row offset 8

    const float* Xb   = X + (size_t)b * NPTS * NDIM;
    const float* arow = Xb + (size_t)(ti * 16 + r) * NDIM + 2 * h; // A: M=r, K pair
    const float* brow = Xb + (size_t)(tj * 16 + r) * NDIM + 2 * h; // B: N=r, K pair

    v8f acc = {};
#pragma unroll
    for (int k0 = 0; k0 < NDIM; k0 += 4) {
        v2f a = *(const v2f*)(arow + k0);
        v2f bb = *(const v2f*)(brow + k0);
        // D = A x B + C  (16x16x4 fp32)
        acc = __builtin_amdgcn_wmma_f32_16x16x4_f32(
            /*neg_a=*/false, a, /*neg_b=*/false, bb,
            /*c_mod=*/(short)0, acc, /*reuse_a=*/false, /*reuse_b=*/false);
    }

    const float* sqb = sq + (size_t)b * NPTS;
    float sqc = sqb[tj * 16 + r];
    float* Dbase = D + (size_t)b * NPTS * NPTS;
#pragma unroll
    for (int v = 0; v < 8; ++v) {
        int row = ti * 16 + v + 8 * h;          // C/D layout: M = v + 8*(lane/16)
        float d2 = sqb[row] + sqc - 2.0f * acc[v];
        d2 = fmaxf(d2, 0.0f);
        Dbase[(size_t)row * NPTS + tj * 16 + r] = sqrtf(d2);
    }
}

// ---------------------------------------------------------------------------
// Kernel 3: Prim's MST. One block (1024 threads = 32 waves) per batch.
// mind/src/visited live in LDS; argmin via wave32 shuffles + LDS combine.
// ---------------------------------------------------------------------------
__global__ __launch_bounds__(1024) void mst_kernel(const float* __restrict__ D,
                                                   int2* __restrict__ pairs) {
    __shared__ float mind[NPTS];
    __shared__ int   srcv[NPTS];
    __shared__ unsigned char visited[NPTS];
    __shared__ float redV[32];
    __shared__ int   redI[32];
    __shared__ int   sh_j;

    const int b   = blockIdx.x;
    const int tid = threadIdx.x;
    const float INF = __builtin_inff();
    const float* Db = D + (size_t)b * NPTS * NPTS;

    for (int i = tid; i < NPTS; i += 1024) {
        visited[i] = (i == 0) ? 1 : 0;
        mind[i]    = (i == 0) ? INF : Db[i];   // row 0
        srcv[i]    = 0;
    }
    __syncthreads();

    for (int step = 0; step < NEDGE; ++step) {
        // ---- argmin(mind), lowest index on ties ----
        float bv = INF;
        int   bi = NPTS;
        for (int i = tid; i < NPTS; i += 1024) {
            float v = mind[i];
            if (v < bv || (v == bv && i < bi)) { bv = v; bi = i; }
        }
#pragma unroll
        for (int off = 16; off > 0; off >>= 1) {
            float ov = __shfl_down(bv, off);
            int   oi = __shfl_down(bi, off);
            if (ov < bv || (ov == bv && oi < bi)) { bv = ov; bi = oi; }
        }
        if ((tid & 31) == 0) { redV[tid >> 5] = bv; redI[tid >> 5] = bi; }
        __syncthreads();
        if (tid < 32) {
            bv = redV[tid]; bi = redI[tid];
#pragma unroll
            for (int off = 16; off > 0; off >>= 1) {
                float ov = __shfl_down(bv, off);
                int   oi = __shfl_down(bi, off);
                if (ov < bv || (ov == bv && oi < bi)) { bv = ov; bi = oi; }
            }
            if (tid == 0) {
                sh_j = bi;
                pairs[(size_t)b * NEDGE + step] = make_int2(srcv[bi], bi);
                visited[bi] = 1;
            }
        }
        __syncthreads();

        // ---- relax with row D[j,:] ----
        const int j = sh_j;
        const float* Drow = Db + (size_t)j * NPTS;
        for (int i = tid; i < NPTS; i += 1024) {
            float dn = Drow[i];
            bool vis = visited[i] != 0;
            if (!vis && dn < mind[i]) { mind[i] = dn; srcv[i] = j; }
            if (vis) mind[i] = INF;
        }
        __syncthreads();
    }
}

// ---------------------------------------------------------------------------
// Kernel 4: zero the output accumulator
// ---------------------------------------------------------------------------
__global__ void zero_kernel(float* out, int n) {
    int i = blockIdx.x * blockDim.x + threadIdx.x;
    if (i < n) out[i] = 0.0f;
}

// ---------------------------------------------------------------------------
// Kernel 5: loss accumulation. One wave per edge (both edge sets).
// dist recomputed from raw inputs: sqrt(sum_d (x_i - x_j)^2)
// ---------------------------------------------------------------------------
__global__ __launch_bounds__(256) void loss_kernel(const float* __restrict__ gts,
                                                   const float* __restrict__ z,
                                                   const int2* __restrict__ px,
                                                   const int2* __restrict__ pz,
                                                   float* __restrict__ out) {
    int gw   = (blockIdx.x * 256 + threadIdx.x) >> 5;
    int lane = threadIdx.x & 31;
    const int total = BATCH * 2 * NEDGE;
    if (gw >= total) return;

    int b = gw / (2 * NEDGE);
    int r = gw % (2 * NEDGE);
    int2 p = (r < NEDGE) ? px[(size_t)b * NEDGE + r]
                         : pz[(size_t)b * NEDGE + (r - NEDGE)];

    const float* xi = gts + ((size_t)b * NPTS + p.x) * NDIM;
    const float* xj = gts + ((size_t)b * NPTS + p.y) * NDIM;
    const float* zi = z   + ((size_t)b * NPTS + p.x) * NDIM;
    const float* zj = z   + ((size_t)b * NPTS + p.y) * NDIM;

    float dx2 = 0.0f, dz2 = 0.0f;
#pragma unroll
    for (int d = lane; d < NDIM; d += 32) {
        float xd = xi[d] - xj[d];
        float zd = zi[d] - zj[d];
        dx2 += xd * xd;
        dz2 += zd * zd;
    }
#pragma unroll
    for (int off = 16; off > 0; off >>= 1) {
        dx2 += __shfl_down(dx2, off);
        dz2 += __shfl_down(dz2, off);
    }
    if (lane == 0) {
        float diff = sqrtf(dx2) - sqrtf(dz2);
        atomicAdd(out, 0.5f * diff * diff);   // TOPO_LAMBDA = 1.0
    }
}

// ---------------------------------------------------------------------------
extern "C" void kernel_launch(void* const* d_in, const int* in_sizes, int n_in,
                              void* d_out, int out_size, void* d_ws, size_t ws_size,
                              hipStream_t stream) {
    const float* gts = (const float*)d_in[0];
    const float* z   = (const float*)d_in[1];
    float* out = (float*)d_out;

    // workspace layout: sq | pairs_x | pairs_z | D (reused for x then z)
    char* ws = (char*)d_ws;
    size_t off = 0;
    float* sq = (float*)(ws + off);
    off += ((size_t)BATCH * NPTS * sizeof(float) + 255) & ~(size_t)255;
    int2* px = (int2*)(ws + off);
    off += ((size_t)BATCH * NEDGE * sizeof(int2) + 255) & ~(size_t)255;
    int2* pz = (int2*)(ws + off);
    off += ((size_t)BATCH * NEDGE * sizeof(int2) + 255) & ~(size_t)255;
    float* D = (float*)(ws + off);   // BATCH*NPTS*NPTS floats = 256 MB

    const int sqBlocks   = (BATCH * NPTS + 255) / 256;
    const int tileWaves  = BATCH * (NPTS / 16) * (NPTS / 16);  // 262144 waves
    const int gramBlocks = tileWaves / 8;                       // 8 waves/block
    const int edgeWaves  = BATCH * 2 * NEDGE;
    const int lossBlocks = (edgeWaves + 7) / 8;

    // ---- phase X ----
    sqnorm_kernel  <<<sqBlocks,   256, 0, stream>>>(gts, sq);
    gram_dist_kernel<<<gramBlocks, 256, 0, stream>>>(gts, sq, D);
    mst_kernel     <<<BATCH,     1024, 0, stream>>>(D, px);
    // ---- phase Z (reuse D buffer) ----
    sqnorm_kernel  <<<sqBlocks,   256, 0, stream>>>(z, sq);
    gram_dist_kernel<<<gramBlocks, 256, 0, stream>>>(z, sq, D);
    mst_kernel     <<<BATCH,     1024, 0, stream>>>(D, pz);
    // ---- loss ----
    zero_kernel    <<<(out_size + 255) / 256, 256, 0, stream>>>(out, out_size);
    loss_kernel    <<<lossBlocks, 256, 0, stream>>>(gts, z, px, pz, out);
}